// APPNALayer_25142738550921
// MI455X (gfx1250) — compile-verified
//
#include <hip/hip_runtime.h>
#include <hip/hip_bf16.h>

#define IN_DIM   128
#define HID_DIM  256
#define OUT_DIM  128
#define KHOPS    10
#define ALPHA    0.1f

typedef __attribute__((ext_vector_type(2))) float v2f;
typedef __attribute__((ext_vector_type(8))) float v8f;

// ---------------- degree / norm ----------------

__global__ void init_deg_kernel(unsigned* __restrict__ deg, int N) {
    int i = blockIdx.x * blockDim.x + threadIdx.x;
    if (i < N) deg[i] = 1u;  // self-loop contributes 1 to every in-degree
}

__global__ void count_deg_kernel(const int* __restrict__ dst, unsigned* __restrict__ deg, int E) {
    int e = blockIdx.x * blockDim.x + threadIdx.x;
    if (e < E) atomicAdd(&deg[dst[e]], 1u);
}

__global__ void norm_kernel(const unsigned* __restrict__ deg, float* __restrict__ norm, int N) {
    int i = blockIdx.x * blockDim.x + threadIdx.x;
    if (i < N) {
        unsigned d = deg[i];
        norm[i] = rsqrtf((float)(d < 1u ? 1u : d));
    }
}

// ---------------- propagation ----------------

// agg[i][:] = h[i][:] * norm[i]   (self-loop term; also zero-initializes accumulator)
__global__ void init_agg_kernel(const float* __restrict__ h, const float* __restrict__ norm,
                                float* __restrict__ agg, int n128) {
    int idx = blockIdx.x * blockDim.x + threadIdx.x;
    if (idx < n128) {
        int i = idx >> 7;  // IN_DIM == 128
        agg[idx] = h[idx] * norm[i];
    }
}

// one wave32 per edge; each lane owns a float4 slice of the 128-dim feature
__global__ void edge_prop_kernel(const float* __restrict__ h, const float* __restrict__ norm,
                                 const int* __restrict__ src, const int* __restrict__ dst,
                                 float* __restrict__ agg, int E) {
    int gid  = blockIdx.x * blockDim.x + threadIdx.x;
    int e    = gid >> 5;
    int lane = gid & 31;
    if (e >= E) return;
    int s = src[e];
    int d = dst[e];
    float ns = norm[s];
    const float4* hp = (const float4*)(h + (size_t)s * IN_DIM);
    float4 v = hp[lane];
    float* ap = agg + (size_t)d * IN_DIM + lane * 4;
    __hip_atomic_fetch_add(ap + 0, v.x * ns, __ATOMIC_RELAXED, __HIP_MEMORY_SCOPE_AGENT);
    __hip_atomic_fetch_add(ap + 1, v.y * ns, __ATOMIC_RELAXED, __HIP_MEMORY_SCOPE_AGENT);
    __hip_atomic_fetch_add(ap + 2, v.z * ns, __ATOMIC_RELAXED, __HIP_MEMORY_SCOPE_AGENT);
    __hip_atomic_fetch_add(ap + 3, v.w * ns, __ATOMIC_RELAXED, __HIP_MEMORY_SCOPE_AGENT);
}

// h_new = (1-alpha) * agg * norm[dst] + alpha * feat
__global__ void combine_kernel(const float* __restrict__ agg, const float* __restrict__ feat,
                               const float* __restrict__ norm, float* __restrict__ hout, int n128) {
    int idx = blockIdx.x * blockDim.x + threadIdx.x;
    if (idx < n128) {
        int i = idx >> 7;
        hout[idx] = (1.0f - ALPHA) * agg[idx] * norm[i] + ALPHA * feat[idx];
    }
}

// ---------------- fused 2-layer MLP via V_WMMA_F32_16X16X4_F32 ----------------
// One wave computes a 16-row tile: relu(H@W1+b1)@W2+b2, hidden tile staged in LDS.
// A frag (16x4 f32): lane L -> row (L&15), K = 2*(L>>4)+{0,1}
// B frag (4x16 f32): lane L -> col (L&15), K = 2*(L>>4)+{0,1}
// C/D (16x16 f32):   VGPR v -> row v + 8*(L>>4), col (L&15)

__global__ __launch_bounds__(64) void mlp_wmma_kernel(
    const float* __restrict__ H,  const float* __restrict__ W1, const float* __restrict__ b1,
    const float* __restrict__ W2, const float* __restrict__ b2, float* __restrict__ out, int N) {
    __shared__ float hid[2][16 * HID_DIM];

    int wave = threadIdx.x >> 5;
    int lane = threadIdx.x & 31;
    int tile = blockIdx.x * 2 + wave;
    int row0 = tile * 16;
    bool active = (row0 < N);

    int m  = lane & 15;          // A-row / B-col / C-col owned by this lane
    int kb = (lane >> 4) * 2;    // K offset within each group of 4
    int hi = (lane >> 4) * 8;    // C/D row offset for upper lane half
    float* hw = &hid[wave][0];

    v2f a1[32];
    if (active) {
        const float* hrow = H + (size_t)(row0 + m) * IN_DIM;
#pragma unroll
        for (int s = 0; s < 32; ++s) {
            a1[s].x = hrow[4 * s + kb];
            a1[s].y = hrow[4 * s + kb + 1];
        }
        // ---- layer 1: hid = relu(H @ W1 + b1), 16 column tiles of 16 ----
#pragma unroll 1
        for (int j = 0; j < HID_DIM / 16; ++j) {
            int col = j * 16 + m;
            v8f c = {};
#pragma unroll
            for (int s = 0; s < 32; ++s) {
                v2f b;
                b.x = W1[(4 * s + kb) * HID_DIM + col];
                b.y = W1[(4 * s + kb + 1) * HID_DIM + col];
                c = __builtin_amdgcn_wmma_f32_16x16x4_f32(false, a1[s], false, b,
                                                          (short)0, c, false, false);
            }
            float bias = b1[col];
#pragma unroll
            for (int v = 0; v < 8; ++v) {
                float val = c[v] + bias;
                hw[(v + hi) * HID_DIM + col] = val > 0.0f ? val : 0.0f;
            }
        }
    }
    __syncthreads();
    if (active) {
        // ---- layer 2: out = hid @ W2 + b2, 8 column tiles of 16 ----
#pragma unroll 1
        for (int jo = 0; jo < OUT_DIM / 16; ++jo) {
            int col = jo * 16 + m;
            v8f c = {};
#pragma unroll 4
            for (int s = 0; s < 64; ++s) {
                v2f a, b;
                a.x = hw[m * HID_DIM + 4 * s + kb];
                a.y = hw[m * HID_DIM + 4 * s + kb + 1];
                b.x = W2[(4 * s + kb) * OUT_DIM + col];
                b.y = W2[(4 * s + kb + 1) * OUT_DIM + col];
                c = __builtin_amdgcn_wmma_f32_16x16x4_f32(false, a, false, b,
                                                          (short)0, c, false, false);
            }
            float bias = b2[col];
#pragma unroll
            for (int v = 0; v < 8; ++v)
                out[(size_t)(row0 + v + hi) * OUT_DIM + col] = c[v] + bias;
        }
    }
}

// ---------------- host launch ----------------

extern "C" void kernel_launch(void* const* d_in, const int* in_sizes, int n_in,
                              void* d_out, int out_size, void* d_ws, size_t ws_size,
                              hipStream_t stream) {
    const float* feat = (const float*)d_in[0];
    const float* W1   = (const float*)d_in[1];
    const float* b1   = (const float*)d_in[2];
    const float* W2   = (const float*)d_in[3];
    const float* b2   = (const float*)d_in[4];
    const int*   src  = (const int*)d_in[5];
    const int*   dst  = (const int*)d_in[6];
    float* out = (float*)d_out;

    int N = in_sizes[0] / IN_DIM;
    int E = in_sizes[5];
    int n128 = N * IN_DIM;

    // workspace layout (all 256B aligned): deg | norm | agg | h
    char* ws = (char*)d_ws;
    size_t o = 0;
    auto align256 = [](size_t x) { return (x + 255) & ~(size_t)255; };
    unsigned* deg  = (unsigned*)(ws + o); o += align256((size_t)N * 4);
    float*    norm = (float*)(ws + o);    o += align256((size_t)N * 4);
    float*    agg  = (float*)(ws + o);    o += align256((size_t)n128 * 4);
    float*    h    = (float*)(ws + o);    o += align256((size_t)n128 * 4);

    const int TB = 256;
    int nb    = (N + TB - 1) / TB;
    int eb    = (E + TB - 1) / TB;
    int n128b = (n128 + TB - 1) / TB;
    long long ethreads = (long long)E * 32;
    int ewaveb = (int)((ethreads + TB - 1) / TB);

    // degrees (with self-loops) -> symmetric norm
    init_deg_kernel<<<nb, TB, 0, stream>>>(deg, N);
    count_deg_kernel<<<eb, TB, 0, stream>>>(dst, deg, E);
    norm_kernel<<<nb, TB, 0, stream>>>(deg, norm, N);

    // K APPNP hops
    const float* hcur = feat;
    for (int t = 0; t < KHOPS; ++t) {
        init_agg_kernel<<<n128b, TB, 0, stream>>>(hcur, norm, agg, n128);
        edge_prop_kernel<<<ewaveb, TB, 0, stream>>>(hcur, norm, src, dst, agg, E);
        combine_kernel<<<n128b, TB, 0, stream>>>(agg, feat, norm, h, n128);
        hcur = h;
    }

    // fused MLP: 16-row tiles, 2 waves per block
    int tiles  = (N + 15) / 16;
    int blocks = (tiles + 1) / 2;
    mlp_wmma_kernel<<<blocks, 64, 0, stream>>>(hcur, W1, b1, W2, b2, out, N);
}